// GraphAttention_66262755442758
// MI455X (gfx1250) — compile-verified
//
#include <hip/hip_runtime.h>
#include <hip/hip_bf16.h>

#define B_ 2
#define N_ 1024
#define D_ 64
#define H_ 64

typedef __attribute__((ext_vector_type(2))) float v2f;
typedef __attribute__((ext_vector_type(8))) float v8f;
typedef int v4i_vs __attribute__((vector_size(16)));

#if defined(__AMDGCN__) && __has_builtin(__builtin_amdgcn_global_load_async_to_lds_b128)
#define HAVE_ASYNC_LDS 1
// builtin signature (from hipcc diagnostic): (v4i as1*, v4i as3*, Ii offset, Ii cpol)
#define ASYNC_G2L_B128(src, dst)                                             \
  __builtin_amdgcn_global_load_async_to_lds_b128(                            \
      (v4i_vs __attribute__((address_space(1)))*)(src),                      \
      (v4i_vs __attribute__((address_space(3)))*)(dst), 0, 0)
#else
#define HAVE_ASYNC_LDS 0
#endif

// Drain ASYNCcnt: prefer the builtin; if the toolchain lacks it but we used the
// async path, fall back to raw inline asm so the wait is never silently dropped
// (s_barrier alone does NOT drain ASYNCcnt).
#if defined(__AMDGCN__) && __has_builtin(__builtin_amdgcn_s_wait_asynccnt)
#define WAIT_ASYNC() __builtin_amdgcn_s_wait_asynccnt(0)
#elif HAVE_ASYNC_LDS
#define WAIT_ASYNC() asm volatile("s_wait_asynccnt 0x0" ::: "memory")
#else
#define WAIT_ASYNC() ((void)0)
#endif

// ---------------------------------------------------------------------------
// Kernel 1: projections.
//   pb[b,n,h] = nodes[b,n,:] . (W1+W3)[:,h] + bias[h]
//   qT[b,h,n] = nodes[b,n,:] . (W2-W3)[:,h]   (transposed for coalesced reads)
// grid: B*N blocks of 64 threads (thread = h)
// ---------------------------------------------------------------------------
__global__ __launch_bounds__(64)
void k_proj(const float* __restrict__ nodes, const float* __restrict__ W,
            const float* __restrict__ bias,
            float* __restrict__ pb, float* __restrict__ qT) {
  const int r = blockIdx.x;        // global row: b*N + n
  const int h = threadIdx.x;       // 0..63
  __shared__ float nd[D_];
  nd[h] = nodes[r * D_ + h];
  __syncthreads();
  float accp = 0.f, accq = 0.f;
#pragma unroll
  for (int d = 0; d < D_; ++d) {
    const float w1 = W[d * H_ + h];
    const float w2 = W[(D_ + d) * H_ + h];
    const float w3 = W[(2 * D_ + d) * H_ + h];
    accp = fmaf(nd[d], w1 + w3, accp);
    accq = fmaf(nd[d], w2 - w3, accq);
  }
  pb[r * H_ + h] = accp + bias[h];
  const int b = r >> 10;
  const int n = r & (N_ - 1);
  qT[((size_t)b * H_ + h) * N_ + n] = accq;
}

// ---------------------------------------------------------------------------
// Kernel 2: pairwise scores (VALU-bound; lrelu(x) = 0.6x + 0.4|x|).
//   S[b,i,j] = 0.125 * sum_h a[h] * lrelu(pb[b,i,h] + qT[b,h,j])
// grid: (N/256, N/16, B), 256 threads; each lane caches q[j][0..63] in VGPRs
// and reuses them across the 16 i-rows of the tile.
// ---------------------------------------------------------------------------
__global__ __launch_bounds__(256)
void k_scores(const float* __restrict__ pb, const float* __restrict__ qT,
              const float* __restrict__ avec, float* __restrict__ S) {
  const int b  = blockIdx.z;
  const int i0 = blockIdx.y * 16;
  const int j  = blockIdx.x * 256 + threadIdx.x;

  __shared__ float pb_s[16 * H_];
  __shared__ float a_s[H_];
  for (int t = threadIdx.x; t < 16 * H_; t += 256)
    pb_s[t] = pb[((size_t)b * N_ + i0) * H_ + t];
  if (threadIdx.x < H_) a_s[threadIdx.x] = avec[threadIdx.x];
  __syncthreads();

  float qv[H_];
  const float* qb = qT + (size_t)b * H_ * N_ + j;
#pragma unroll
  for (int h = 0; h < H_; ++h) qv[h] = qb[(size_t)h * N_];

  float* Srow = S + (size_t)b * N_ * N_ + (size_t)i0 * N_ + j;
  for (int i = 0; i < 16; ++i) {
    float acc = 0.f;
#pragma unroll
    for (int h = 0; h < H_; ++h) {
      const float t  = pb_s[i * H_ + h] + qv[h];
      const float lr = fmaf(0.4f, fabsf(t), 0.6f * t);   // leaky-relu(0.2)
      acc = fmaf(a_s[h], lr, acc);
    }
    Srow[(size_t)i * N_] = acc * 0.125f;                 // 1/sqrt(D)
  }
}

// ---------------------------------------------------------------------------
// Kernel 3: row softmax, one wave32 per row, normalized in place.
// ---------------------------------------------------------------------------
__global__ __launch_bounds__(32)
void k_softmax(float* __restrict__ S) {
  float* row = S + (size_t)blockIdx.x * N_;
  const int lane = threadIdx.x;

  float m = -3.402823466e38f;
  for (int jj = lane; jj < N_; jj += 32) m = fmaxf(m, row[jj]);
#pragma unroll
  for (int off = 16; off > 0; off >>= 1) m = fmaxf(m, __shfl_xor(m, off, 32));

  float s = 0.f;
  for (int jj = lane; jj < N_; jj += 32) {
    const float e = __expf(row[jj] - m);
    row[jj] = e;
    s += e;
  }
#pragma unroll
  for (int off = 16; off > 0; off >>= 1) s += __shfl_xor(s, off, 32);

  const float inv = 1.0f / s;
  for (int jj = lane; jj < N_; jj += 32) row[jj] *= inv;
}

// ---------------------------------------------------------------------------
// Kernel 4: out = att @ nodes via V_WMMA_F32_16X16X4_F32, with the 16-row
// attention slab staged into LDS through the CDNA5 async global->LDS path.
//
// K is processed in two 512-wide chunks. LDS pitch = 516 floats:
//   - 516 % 64 == 4  -> lanes 0-15 (stride 516 words) hit 16 distinct banks
//   - lane-half k-offset (+2 words) lands on the disjoint bank residues
//   - every A pair (even k) is 8-byte aligned for ds_load_b64
//
// A (16x4 f32): lanes 0-15 M=lane%16, VGPR0=K(2*half), VGPR1=K(2*half)+1.
// B (4x16 f32): lanes 0-15 N=lane%16, VGPR0=K(2*half), VGPR1=K(2*half)+1.
// C (16x16 f32): M = r + 8*half, N = lane%16.
// grid: (N/16, B), 128 threads = 4 waves; wave w owns output col-tile n0=16w.
// ---------------------------------------------------------------------------
#define KC_   512                  // K-chunk width
#define PITCH 516                  // LDS row pitch in floats (512 + 4)

__global__ __launch_bounds__(128)
void k_av(const float* __restrict__ S, const float* __restrict__ nodes,
          float* __restrict__ out) {
  const int b    = blockIdx.y;
  const int i0   = blockIdx.x * 16;
  const int tid  = threadIdx.x;
  const int wave = tid >> 5;
  const int lane = tid & 31;
  const int half = lane >> 4;
  const int lm   = lane & 15;
  const int n0   = wave * 16;

  const float* Wt = S + (size_t)b * N_ * N_ + (size_t)i0 * N_;   // 16 x 1024
  const float* Nb = nodes + (size_t)b * N_ * D_;                 // 1024 x 64

  __shared__ float smemA[16 * PITCH];                            // 33 KB

  v8f c = {};
  for (int kc = 0; kc < N_; kc += KC_) {
    // ---- stage A chunk (16 x 512 f32) into LDS asynchronously -------------
    // 2048 b128 chunks, 16 per thread.
#pragma unroll
    for (int t = 0; t < 16; ++t) {
      const int id  = t * 128 + tid;
      const int row = id >> 7;          // 0..15
      const int c4  = id & 127;         // 0..127 (x4 floats)
      const float* src = Wt + (size_t)row * N_ + kc + c4 * 4;
      float* dst = &smemA[row * PITCH + c4 * 4];
#if HAVE_ASYNC_LDS
      ASYNC_G2L_B128(src, dst);
#else
      *(float4*)dst = *(const float4*)src;
#endif
    }
    WAIT_ASYNC();
    __syncthreads();

    // ---- WMMA over this K chunk ------------------------------------------
    for (int k0 = 0; k0 < KC_; k0 += 4) {
      const int kl = k0 + 2 * half;           // lane-half K offset (even)
      const int kg = kc + kl;                 // global K
      v2f A = *(const v2f*)&smemA[lm * PITCH + kl];
      v2f Bv;
      Bv.x = Nb[(size_t)kg * D_ + n0 + lm];
      Bv.y = Nb[(size_t)(kg + 1) * D_ + n0 + lm];
      c = __builtin_amdgcn_wmma_f32_16x16x4_f32(
          /*neg_a=*/false, A, /*neg_b=*/false, Bv,
          /*c_mod=*/(short)0, c, /*reuse_a=*/false, /*reuse_b=*/false);
    }
    __syncthreads();                          // chunk fully consumed
  }

  float* ob = out + ((size_t)b * N_ + i0) * D_ + n0 + lm;
#pragma unroll
  for (int r = 0; r < 8; ++r) {
    const int m = r + 8 * half;
    ob[(size_t)m * D_] = c[r];
  }
}

// ---------------------------------------------------------------------------
extern "C" void kernel_launch(void* const* d_in, const int* in_sizes, int n_in,
                              void* d_out, int out_size, void* d_ws, size_t ws_size,
                              hipStream_t stream) {
  const float* nodes = (const float*)d_in[0];   // (B, N, D)
  const float* W     = (const float*)d_in[1];   // (3D, H)
  const float* bias  = (const float*)d_in[2];   // (H,)
  const float* avec  = (const float*)d_in[3];   // (H,)
  float* outp = (float*)d_out;                  // (B, N, D)

  float* ws = (float*)d_ws;
  float* pb = ws;                               // B*N*H   = 131072 floats
  float* qT = pb + (size_t)B_ * N_ * H_;        // B*H*N   = 131072 floats
  float* S  = qT + (size_t)B_ * H_ * N_;        // B*N*N   = 2097152 floats

  k_proj<<<dim3(B_ * N_), dim3(64), 0, stream>>>(nodes, W, bias, pb, qT);
  k_scores<<<dim3(N_ / 256, N_ / 16, B_), dim3(256), 0, stream>>>(pb, qT, avec, S);
  k_softmax<<<dim3(B_ * N_), dim3(32), 0, stream>>>(S);
  k_av<<<dim3(N_ / 16, B_), dim3(128), 0, stream>>>(S, nodes, outp);
}